// GNNVariationalExample_4406636445741
// MI455X (gfx1250) — compile-verified
//
#include <hip/hip_runtime.h>
#include <math.h>

// ---------------- problem constants (match reference) ----------------
#define Gg      16
#define Nn      2048
#define Ff      32
#define Ee      32768
#define IN_DIM  65536      // N*F
#define Hh      512
#define Ll      64
#define BN_EPS  1e-5f
#define SLOPE   0.01f

typedef float v2f __attribute__((ext_vector_type(2)));
typedef float v8f __attribute__((ext_vector_type(8)));

__device__ __forceinline__ float atomAddF(float* p, float v) {
    return __hip_atomic_fetch_add(p, v, __ATOMIC_RELAXED, __HIP_MEMORY_SCOPE_AGENT);
}

// ---------------- fp32 WMMA 16x16xK inner loop ----------------
// A: 16 x K tile base (row 0, k 0), row stride lda (floats)
// B: K x 16 tile base (k 0, col 0), row stride ldb (floats)
// Layout per CDNA5 ISA 7.12.2 (32-bit matrices, wave32):
//   A: lane m = lane&15; VGPR0/1 hold K = 2*(lane>>4) + {0,1}  -> float2 load
//   B: lane n = lane&15; VGPR0/1 hold K = 2*(lane>>4) + {0,1}
__device__ __forceinline__ v8f wmma_f32_loop(const float* __restrict__ A, int lda,
                                             const float* __restrict__ B, int ldb,
                                             int K, int lane, v8f c) {
    const int mn = lane & 15;
    const int kb = (lane >> 4) * 2;
    for (int k = 0; k < K; k += 4) {
        v2f a = *(const v2f*)(A + (size_t)mn * lda + (k + kb));
        v2f b;
        b.x = B[(size_t)(k + kb)     * ldb + mn];
        b.y = B[(size_t)(k + kb + 1) * ldb + mn];
        c = __builtin_amdgcn_wmma_f32_16x16x4_f32(false, a, false, b,
                                                  (short)0, c, false, false);
    }
    return c;
}

// ---------------- GCN stage ----------------
__global__ void k_init_deg(float* deg) {
    int i = blockIdx.x * 256 + threadIdx.x;
    if (i < Gg * Nn) deg[i] = 1.0f;            // self loop
}

__global__ void k_deg_scatter(const int* __restrict__ ei, float* deg) {
    int i = blockIdx.x * 256 + threadIdx.x;    // over G*E
    if (i >= Gg * Ee) return;
    int g = i / Ee, e = i - g * Ee;
    int dst = ei[(size_t)(g * 2 + 1) * Ee + e];
    atomAddF(&deg[g * Nn + dst], 1.0f);
}

__global__ void k_dinv(const float* __restrict__ deg, float* dinv) {
    int i = blockIdx.x * 256 + threadIdx.x;
    if (i < Gg * Nn) dinv[i] = rsqrtf(deg[i]);
}

// xw[g] = x[g] @ W_gcn   ([2048,32] @ [32,32]) -- WMMA tiles
__global__ void k_gemm_xw(const float* __restrict__ x, const float* __restrict__ Wg,
                          float* __restrict__ xw) {
    int blk = blockIdx.x;             // G * 128 * 2 = 4096 single-wave blocks
    int ct  = blk & 1;
    int rt  = (blk >> 1) & 127;
    int g   = blk >> 8;
    int lane = threadIdx.x;
    const float* A = x  + (size_t)(g * Nn + rt * 16) * Ff;
    const float* B = Wg + ct * 16;
    v8f c = {};
    c = wmma_f32_loop(A, Ff, B, Ff, Ff, lane, c);
    int n = lane & 15, hi = lane >> 4;
    float* D = xw + (size_t)(g * Nn + rt * 16) * Ff + ct * 16;
    #pragma unroll
    for (int r = 0; r < 8; ++r)
        D[(size_t)(r + 8 * hi) * Ff + n] = c[r];
}

// self-loop contribution: h = xw * dinv^2 = xw / deg
__global__ void k_init_h(const float* __restrict__ xw, const float* __restrict__ deg,
                         float* __restrict__ h) {
    int i = blockIdx.x * 256 + threadIdx.x;    // over G*N*F
    if (i >= Gg * Nn * Ff) return;
    h[i] = xw[i] / deg[i >> 5];                // F == 32
}

// one wave per edge, lane = feature: h[dst] += xw[src] * dinv[src]*dinv[dst]
__global__ void k_edge_scatter(const int* __restrict__ ei, const float* __restrict__ xw,
                               const float* __restrict__ dinv, float* __restrict__ h) {
    int ge   = blockIdx.x * 8 + (threadIdx.x >> 5);   // global edge id
    int lane = threadIdx.x & 31;
    if (ge >= Gg * Ee) return;
    int g = ge / Ee, e = ge - g * Ee;
    int src = ei[(size_t)(g * 2)     * Ee + e];
    int dst = ei[(size_t)(g * 2 + 1) * Ee + e];
    float coef = dinv[g * Nn + src] * dinv[g * Nn + dst];
    float v = xw[(size_t)(g * Nn + src) * Ff + lane] * coef;
    atomAddF(&h[(size_t)(g * Nn + dst) * Ff + lane], v);
}

// ---------------- BatchNorm(batch stats over nodes) + tanh -> xs ----------------
// one block (256 thr = 8 waves) per graph; lane = feature, waves split nodes
__global__ void k_bn_tanh(const float* __restrict__ h, const float* __restrict__ b_gcn,
                          const float* __restrict__ gamma, const float* __restrict__ beta,
                          float* __restrict__ xs) {
    __shared__ float s_sum[8][32];
    __shared__ float s_sq [8][32];
    __shared__ float s_mean[32];
    __shared__ float s_rstd[32];
    int g    = blockIdx.x;
    int lane = threadIdx.x & 31;
    int w    = threadIdx.x >> 5;
    const float bias = b_gcn[lane];
    const float* hg = h + (size_t)g * Nn * Ff;

    float sum = 0.f, sq = 0.f;
    for (int n = w * 256; n < (w + 1) * 256; ++n) {
        float v = hg[(size_t)n * Ff + lane] + bias;
        sum += v; sq += v * v;
    }
    s_sum[w][lane] = sum; s_sq[w][lane] = sq;
    __syncthreads();
    if (threadIdx.x < 32) {
        float ts = 0.f, tq = 0.f;
        #pragma unroll
        for (int i = 0; i < 8; ++i) { ts += s_sum[i][lane]; tq += s_sq[i][lane]; }
        float mean = ts * (1.0f / Nn);
        float var  = tq * (1.0f / Nn) - mean * mean;   // biased, as torch BN
        s_mean[lane] = mean;
        s_rstd[lane] = rsqrtf(var + BN_EPS);
    }
    __syncthreads();
    float mean = s_mean[lane], rstd = s_rstd[lane];
    float gm = gamma[lane], bt = beta[lane];
    float* xg = xs + (size_t)g * IN_DIM;
    for (int n = w * 256; n < (w + 1) * 256; ++n) {
        float v = hg[(size_t)n * Ff + lane] + bias;
        xg[(size_t)n * Ff + lane] = tanhf((v - mean) * rstd * gm + bt);
    }
}

// ---------------- VAE encoder trunk: henc_pre = xs @ W_enc (K-split) ----------------
__global__ void k_zero(float* p, int n) {
    int i = blockIdx.x * 256 + threadIdx.x;
    if (i < n) p[i] = 0.f;
}

__global__ void k_gemm_enc(const float* __restrict__ xs, const float* __restrict__ W,
                           float* __restrict__ accum) {
    int ct = blockIdx.x;                  // 0..31  (col tile of H=512)
    int kc = blockIdx.y;                  // 0..127 (K chunk of 512)
    int lane = threadIdx.x;
    const float* A = xs + (size_t)kc * 512;                      // lda = IN_DIM
    const float* B = W + (size_t)(kc * 512) * Hh + ct * 16;      // ldb = H
    v8f c = {};
    c = wmma_f32_loop(A, IN_DIM, B, Hh, 512, lane, c);
    int n = lane & 15, hi = lane >> 4;
    float* D = accum + ct * 16;
    #pragma unroll
    for (int r = 0; r < 8; ++r)
        atomAddF(&D[(size_t)(r + 8 * hi) * Hh + n], c[r]);
}

__global__ void k_bias_lrelu(const float* __restrict__ pre, const float* __restrict__ b,
                             float* __restrict__ out, int cols, int n) {
    int i = blockIdx.x * 256 + threadIdx.x;
    if (i >= n) return;
    float v = pre[i] + b[i % cols];
    out[i] = v > 0.f ? v : v * SLOPE;
}

// ---------------- heads: mean / log_var = henc @ W + b ----------------
__global__ void k_gemm_head(const float* __restrict__ henc,
                            const float* __restrict__ Wmu, const float* __restrict__ bmu,
                            const float* __restrict__ Wlv, const float* __restrict__ blv,
                            float* __restrict__ o_mean, float* __restrict__ o_lv) {
    int blk = blockIdx.x;                 // 0..7 : [mu x4 tiles | lv x4 tiles]
    int mat = blk >> 2, ct = blk & 3;
    const float* W = mat ? Wlv : Wmu;
    const float* bb = mat ? blv : bmu;
    float* O = mat ? o_lv : o_mean;
    int lane = threadIdx.x;
    v8f c = {};
    c = wmma_f32_loop(henc, Hh, W + ct * 16, Ll, Hh, lane, c);
    int n = lane & 15, hi = lane >> 4;
    #pragma unroll
    for (int r = 0; r < 8; ++r)
        O[(size_t)(r + 8 * hi) * Ll + ct * 16 + n] = c[r] + bb[ct * 16 + n];
}

__global__ void k_reparam(const float* __restrict__ mean, const float* __restrict__ lv,
                          const float* __restrict__ eps, float* __restrict__ z) {
    int i = blockIdx.x * 256 + threadIdx.x;
    if (i < Gg * Ll) z[i] = mean[i] + expf(0.5f * lv[i]) * eps[i];
}

// ---------------- decoder ----------------
// hd = lrelu(z @ W_dec1 + b_dec1)   [16,64]@[64,512]
__global__ void k_gemm_dec1(const float* __restrict__ z, const float* __restrict__ W,
                            const float* __restrict__ b, float* __restrict__ hd) {
    int ct = blockIdx.x;                  // 0..31
    int lane = threadIdx.x;
    v8f c = {};
    c = wmma_f32_loop(z, Ll, W + ct * 16, Hh, Ll, lane, c);
    int n = lane & 15, hi = lane >> 4;
    #pragma unroll
    for (int r = 0; r < 8; ++r) {
        float v = c[r] + b[ct * 16 + n];
        hd[(size_t)(r + 8 * hi) * Hh + ct * 16 + n] = v > 0.f ? v : v * SLOPE;
    }
}

// x_hat = hd @ W_dec2 + b_dec2      [16,512]@[512,65536]
__global__ void k_gemm_dec2(const float* __restrict__ hd, const float* __restrict__ W,
                            const float* __restrict__ b, float* __restrict__ xhat) {
    int ct = blockIdx.x;                  // 0..4095
    int lane = threadIdx.x;
    v8f c = {};
    c = wmma_f32_loop(hd, Hh, W + (size_t)ct * 16, IN_DIM, Hh, lane, c);
    int n = lane & 15, hi = lane >> 4;
    int col = ct * 16 + n;
    #pragma unroll
    for (int r = 0; r < 8; ++r)
        xhat[(size_t)(r + 8 * hi) * IN_DIM + col] = c[r] + b[col];
}

// ---------------- launcher ----------------
extern "C" void kernel_launch(void* const* d_in, const int* in_sizes, int n_in,
                              void* d_out, int out_size, void* d_ws, size_t ws_size,
                              hipStream_t stream) {
    const float* x       = (const float*)d_in[0];
    const int*   ei      = (const int*)  d_in[1];
    const float* eps     = (const float*)d_in[2];
    const float* W_gcn   = (const float*)d_in[3];
    const float* b_gcn   = (const float*)d_in[4];
    const float* bn_g    = (const float*)d_in[5];
    const float* bn_b    = (const float*)d_in[6];
    const float* W_enc   = (const float*)d_in[7];
    const float* b_enc   = (const float*)d_in[8];
    const float* W_mu    = (const float*)d_in[9];
    const float* b_mu    = (const float*)d_in[10];
    const float* W_lv    = (const float*)d_in[11];
    const float* b_lv    = (const float*)d_in[12];
    const float* W_dec1  = (const float*)d_in[13];
    const float* b_dec1  = (const float*)d_in[14];
    const float* W_dec2  = (const float*)d_in[15];
    const float* b_dec2  = (const float*)d_in[16];

    float* out    = (float*)d_out;
    float* o_xhat = out;                       // 16*65536
    float* o_mean = out + (size_t)Gg * IN_DIM; // 16*64
    float* o_lv   = o_mean + Gg * Ll;          // 16*64

    float* ws       = (float*)d_ws;
    float* xw       = ws;                      ws += (size_t)Gg * Nn * Ff;  // 1M
    float* deg      = ws;                      ws += Gg * Nn;               // 32K
    float* dinv     = ws;                      ws += Gg * Nn;               // 32K
    float* h        = ws;                      ws += (size_t)Gg * Nn * Ff;  // 1M
    float* xs       = ws;                      ws += (size_t)Gg * IN_DIM;   // 1M
    float* henc_pre = ws;                      ws += Gg * Hh;               // 8K
    float* henc     = ws;                      ws += Gg * Hh;               // 8K
    float* z        = ws;                      ws += Gg * Ll;               // 1K
    float* hd       = ws;                      /* 8K */

    // --- GCN conv ---
    k_init_deg    <<<(Gg*Nn + 255)/256, 256, 0, stream>>>(deg);
    k_deg_scatter <<<(Gg*Ee)/256, 256, 0, stream>>>(ei, deg);
    k_dinv        <<<(Gg*Nn + 255)/256, 256, 0, stream>>>(deg, dinv);
    k_gemm_xw     <<<Gg * 128 * 2, 32, 0, stream>>>(x, W_gcn, xw);
    k_init_h      <<<(Gg*Nn*Ff)/256, 256, 0, stream>>>(xw, deg, h);
    k_edge_scatter<<<(Gg*Ee)/8, 256, 0, stream>>>(ei, xw, dinv, h);

    // --- BN + tanh -> xs ---
    k_bn_tanh<<<Gg, 256, 0, stream>>>(h, b_gcn, bn_g, bn_b, xs);

    // --- encoder trunk (K-split over 65536) ---
    k_zero<<<(Gg*Hh + 255)/256, 256, 0, stream>>>(henc_pre, Gg*Hh);
    {
        dim3 grid(Hh / 16, IN_DIM / 512);      // 32 x 128
        k_gemm_enc<<<grid, 32, 0, stream>>>(xs, W_enc, henc_pre);
    }
    k_bias_lrelu<<<(Gg*Hh + 255)/256, 256, 0, stream>>>(henc_pre, b_enc, henc, Hh, Gg*Hh);

    // --- heads + reparameterize ---
    k_gemm_head<<<8, 32, 0, stream>>>(henc, W_mu, b_mu, W_lv, b_lv, o_mean, o_lv);
    k_reparam  <<<(Gg*Ll + 255)/256, 256, 0, stream>>>(o_mean, o_lv, eps, z);

    // --- decoder ---
    k_gemm_dec1<<<Hh / 16, 32, 0, stream>>>(z, W_dec1, b_dec1, hd);
    k_gemm_dec2<<<IN_DIM / 16, 32, 0, stream>>>(hd, W_dec2, b_dec2, o_xhat);
}